// GraphNet_5274219839835
// MI455X (gfx1250) — compile-verified
//
#include <hip/hip_runtime.h>
#include <hip/hip_bf16.h>

typedef float v2f __attribute__((ext_vector_type(2)));
typedef float v8f __attribute__((ext_vector_type(8)));

#define N_NODES 50000
#define E_EDGES 800000
#define DIM     256
#define LDSTR   260   // padded LDS row stride (floats): 260 % 64 == 4 -> conflict-free

// ---------------------------------------------------------------------------
// Dense fp32 GEMM: Hout[16 rows tile] = X_tile(16xD) @ W(DxD), via
// V_WMMA_F32_16X16X4_F32.  One block = 8 waves; wave w covers n-tiles 2w,2w+1.
// ---------------------------------------------------------------------------
__global__ __launch_bounds__(256) void gcn_gemm_wmma(const float* __restrict__ X,
                                                     const float* __restrict__ W,
                                                     float* __restrict__ Hout) {
  __shared__ float ldsA[16 * LDSTR];
  const int m0 = blockIdx.x * 16;
  const int t  = threadIdx.x;

  // Cooperative load of the 16x256 X tile into LDS (coalesced float4s).
  {
    const int row = t >> 4;          // 0..15
    const int cs  = (t & 15) << 4;   // 0,16,...,240
    const float4* g = (const float4*)(X + (size_t)(m0 + row) * DIM + cs);
    float* l = ldsA + row * LDSTR + cs;
    ((float4*)l)[0] = g[0];
    ((float4*)l)[1] = g[1];
    ((float4*)l)[2] = g[2];
    ((float4*)l)[3] = g[3];
  }
  __syncthreads();

  const int lane = t & 31;
  const int wave = t >> 5;       // 0..7
  const int half = lane >> 4;    // 0|1 : K sub-pair selector for A/B frags
  const int l16  = lane & 15;

  const float* Arow = ldsA + l16 * LDSTR;   // A frag: lane holds row M=l16

  for (int nt = wave * 2; nt < wave * 2 + 2; ++nt) {
    const int n0 = nt * 16;
    const float* Wcol = W + n0 + l16;       // B frag: lane holds col N=n0+l16
    v8f c = {};
#pragma unroll 4
    for (int k = 0; k < DIM; k += 4) {
      // A 16x4 fp32 frag: VGPR0/1 = K {0,1} (lanes 0-15) or K {2,3} (lanes 16-31)
      v2f a = *(const v2f*)(Arow + k + 2 * half);
      v2f b;
      b.x = Wcol[(size_t)(k + 2 * half) * DIM];
      b.y = Wcol[(size_t)(k + 2 * half + 1) * DIM];
      c = __builtin_amdgcn_wmma_f32_16x16x4_f32(false, a, false, b,
                                                (short)0, c, false, false);
    }
    // C/D layout: VGPR r -> row (r + 8*half), col l16
    float* out = Hout + (size_t)(m0 + 8 * half) * DIM + n0 + l16;
#pragma unroll
    for (int r = 0; r < 8; ++r) out[(size_t)r * DIM] = c[r];
  }
}

// ---------------------------------------------------------------------------
// Degree / norm precompute
// ---------------------------------------------------------------------------
__global__ void deg_init(float* __restrict__ deg) {
  int i = blockIdx.x * blockDim.x + threadIdx.x;
  if (i < N_NODES) deg[i] = 1.0f;   // self-loop
}

__global__ void deg_accum(const int* __restrict__ dst, float* __restrict__ deg) {
  int e = blockIdx.x * blockDim.x + threadIdx.x;
  if (e < E_EDGES) atomicAdd(&deg[dst[e]], 1.0f);
}

__global__ void edge_norm(const int* __restrict__ src, const int* __restrict__ dst,
                          const float* __restrict__ deg, float* __restrict__ nrm) {
  int e = blockIdx.x * blockDim.x + threadIdx.x;
  if (e < E_EDGES) nrm[e] = __frsqrt_rn(deg[src[e]]) * __frsqrt_rn(deg[dst[e]]);
}

// ---------------------------------------------------------------------------
// agg = h_lin * deg^-1 + bias   (initializes accumulation buffer)
// ---------------------------------------------------------------------------
__global__ void init_agg(const float* __restrict__ h_lin, const float* __restrict__ deg,
                         const float* __restrict__ bias, float* __restrict__ agg) {
  size_t i = (size_t)blockIdx.x * blockDim.x + threadIdx.x;
  if (i < (size_t)N_NODES * DIM) {
    int node = (int)(i >> 8);   // D == 256
    int col  = (int)(i & 255);
    agg[i] = h_lin[i] * (1.0f / deg[node]) + bias[col];
  }
}

// ---------------------------------------------------------------------------
// Edge scatter: agg[dst] += h_lin[src] * norm[e].  64 threads per edge,
// each owning a float4 chunk of the 256-wide feature row (coalesced gather).
// ---------------------------------------------------------------------------
__global__ void edge_scatter(const float* __restrict__ h_lin,
                             const int* __restrict__ src, const int* __restrict__ dst,
                             const float* __restrict__ nrm, float* __restrict__ agg) {
  size_t tid = (size_t)blockIdx.x * blockDim.x + threadIdx.x;
  if (tid >= (size_t)E_EDGES * (DIM / 4)) return;
  int e  = (int)(tid >> 6);          // DIM/4 == 64 chunks per edge
  int c4 = ((int)tid & 63) << 2;     // feature offset
  int s = src[e], d = dst[e];
  float w = nrm[e];
  float4 v = *(const float4*)(h_lin + (size_t)s * DIM + c4);
  float* a = agg + (size_t)d * DIM + c4;
  atomicAdd(a + 0, v.x * w);
  atomicAdd(a + 1, v.y * w);
  atomicAdd(a + 2, v.z * w);
  atomicAdd(a + 3, v.w * w);
}

__global__ void relu_copy(const float* __restrict__ in, float* __restrict__ out) {
  size_t i = (size_t)blockIdx.x * blockDim.x + threadIdx.x;
  if (i < (size_t)N_NODES * DIM) out[i] = fmaxf(in[i], 0.0f);
}

// ---------------------------------------------------------------------------
extern "C" void kernel_launch(void* const* d_in, const int* in_sizes, int n_in,
                              void* d_out, int out_size, void* d_ws, size_t ws_size,
                              hipStream_t stream) {
  const float* x  = (const float*)d_in[0];   // [N, D]
  const float* Ws = (const float*)d_in[1];   // [L, D, D]
  const float* bs = (const float*)d_in[2];   // [L, D]
  const int*   ei = (const int*)d_in[3];     // [2, E]
  const int* src = ei;
  const int* dst = ei + E_EDGES;
  float* out = (float*)d_out;

  const size_t nd = (size_t)N_NODES * DIM;
  float* h_lin = (float*)d_ws;         // N*D
  float* agg   = h_lin + nd;           // N*D
  float* hbuf  = agg + nd;             // N*D (layer output ping buffer)
  float* deg   = hbuf + nd;            // N
  float* nrm   = deg + N_NODES;        // E

  const int B = 256;
  // Structure precompute (degree with self-loop, symmetric edge norms)
  deg_init  <<<(N_NODES + B - 1) / B, B, 0, stream>>>(deg);
  deg_accum <<<(E_EDGES + B - 1) / B, B, 0, stream>>>(dst, deg);
  edge_norm <<<(E_EDGES + B - 1) / B, B, 0, stream>>>(src, dst, deg, nrm);

  const size_t ndThreads = nd;                                  // 12.8M
  const size_t scThreads = (size_t)E_EDGES * (DIM / 4);         // 51.2M
  for (int l = 0; l < 3; ++l) {
    const float* hin = (l == 0) ? x : hbuf;
    const float* W   = Ws + (size_t)l * DIM * DIM;
    const float* b   = bs + (size_t)l * DIM;
    float* hout      = (l == 2) ? out : hbuf;

    gcn_gemm_wmma<<<N_NODES / 16, 256, 0, stream>>>(hin, W, h_lin);
    init_agg     <<<(int)((ndThreads + B - 1) / B), B, 0, stream>>>(h_lin, deg, b, agg);
    edge_scatter <<<(int)((scThreads + B - 1) / B), B, 0, stream>>>(h_lin, src, dst, nrm, agg);
    relu_copy    <<<(int)((ndThreads + B - 1) / B), B, 0, stream>>>(agg, hout);
  }
}